// FFSlotAttentionEncoder_72241349918766
// MI455X (gfx1250) — compile-verified
//
#include <hip/hip_runtime.h>

// FF Slot Attention Encoder for MI455X (gfx1250, wave32).
// B=256, S=2048, D_IN=32, D_SLOT=64, N_HEADS=2, K=8.
// MLP GEMMs run on V_WMMA_F32_16X16X4_F32 (f32 in == reference precision).
// Workspace use: scores (B*S f32, 2 MB) + topk idx (B*K i32, 8 KB)
//                + ctx partials (B*16*64 f32, 1 MB)  => ~3.2 MB total.

typedef __attribute__((ext_vector_type(2))) float v2f;
typedef __attribute__((ext_vector_type(8))) float v8f;

#define B_   256
#define S_   2048
#define DIN  32
#define DSL  64
#define K_   8

#define HS_STRIDE 66  // 64 + 2 pad: keeps 8B alignment, kills LDS bank conflicts

struct Acc4 { v8f a[4]; };

// ---------------------------------------------------------------------------
// Per-wave 16-row MLP tile:  relu(X[16x32] @ W1[32x64] + b1) @ W2[64x64] + b2
// Returns D in WMMA C/D layout: acc.a[nt][v] = Hs2[v + 8*(lane>>4)][16*nt + (lane&15)]
// Weights are staged in LDS pair-interleaved: Wp[(k>>1)*128 + 2n + (k&1)]
// so each 4x16 B-fragment load is a single conflict-free ds_load_b64.
// ---------------------------------------------------------------------------
__device__ __forceinline__ Acc4 mlp_tile16(
    const float* __restrict__ featRows,    // &feats[(b*S + row0)*DIN]
    const float* __restrict__ sW1p, const float* __restrict__ sW2p,
    const float* __restrict__ sB1,  const float* __restrict__ sB2,
    float* __restrict__ hsb)               // per-wave LDS, 16*HS_STRIDE floats
{
  const int lane = threadIdx.x & 31;
  const int h = lane >> 4;   // half-wave: selects K sub-pair (A) / row pair (B)
  const int r = lane & 15;   // A row / B,C,D column-within-tile

  // ---- layer 1: A fragments straight from global (8B-aligned float2) ----
  v2f a1[8];
#pragma unroll
  for (int kc = 0; kc < 8; ++kc)
    a1[kc] = *(const v2f*)(featRows + (size_t)r * DIN + 4 * kc + 2 * h);

  Acc4 acc{};
#pragma unroll
  for (int nt = 0; nt < 4; ++nt) {
#pragma unroll
    for (int kc = 0; kc < 8; ++kc) {
      v2f bf = *(const v2f*)(sW1p + (2 * kc + h) * 128 + (16 * nt + r) * 2);
      acc.a[nt] = __builtin_amdgcn_wmma_f32_16x16x4_f32(
          false, a1[kc], false, bf, (short)0, acc.a[nt], false, false);
    }
  }

  // ---- bias + ReLU, spill D-layout -> row-major LDS tile ----
#pragma unroll
  for (int nt = 0; nt < 4; ++nt) {
    const float bias = sB1[16 * nt + r];
#pragma unroll
    for (int v = 0; v < 8; ++v) {
      float x = acc.a[nt][v] + bias;
      hsb[(v + 8 * h) * HS_STRIDE + 16 * nt + r] = x > 0.f ? x : 0.f;
    }
  }

  // ---- layer 2: reload as A fragments (LDS round trip = layout shuffle) ----
  v2f a2[16];
#pragma unroll
  for (int kc = 0; kc < 16; ++kc)
    a2[kc] = *(const v2f*)(hsb + r * HS_STRIDE + 4 * kc + 2 * h);

  Acc4 acc2{};
#pragma unroll
  for (int nt = 0; nt < 4; ++nt) {
#pragma unroll
    for (int kc = 0; kc < 16; ++kc) {
      v2f bf = *(const v2f*)(sW2p + (2 * kc + h) * 128 + (16 * nt + r) * 2);
      acc2.a[nt] = __builtin_amdgcn_wmma_f32_16x16x4_f32(
          false, a2[kc], false, bf, (short)0, acc2.a[nt], false, false);
    }
  }
#pragma unroll
  for (int nt = 0; nt < 4; ++nt) {
    const float bias = sB2[16 * nt + r];
#pragma unroll
    for (int v = 0; v < 8; ++v) acc2.a[nt][v] += bias;
  }
  return acc2;
}

__device__ __forceinline__ void stage_weights(
    const float* __restrict__ W1, const float* __restrict__ b1,
    const float* __restrict__ W2, const float* __restrict__ b2,
    float* sW1p, float* sW2p, float* sB1, float* sB2)
{
  const int t = threadIdx.x;
  for (int i = t; i < DIN * DSL; i += 256) {
    int k = i >> 6, n = i & 63;
    sW1p[(k >> 1) * 128 + n * 2 + (k & 1)] = W1[i];
  }
  for (int i = t; i < DSL * DSL; i += 256) {
    int k = i >> 6, n = i & 63;
    sW2p[(k >> 1) * 128 + n * 2 + (k & 1)] = W2[i];
  }
  if (t < DSL) { sB1[t] = b1[t]; sB2[t] = b2[t]; }
}

// ---------------------------------------------------------------------------
// Kernel 1: fused MLP + per-row head-averaged scores.
// grid = (S/128, B); 256 threads = 8 waves x 16 rows.
// ---------------------------------------------------------------------------
__global__ __launch_bounds__(256) void k_mlp_scores(
    const float* __restrict__ feats, const float* __restrict__ W1,
    const float* __restrict__ b1,    const float* __restrict__ W2,
    const float* __restrict__ b2,    const float* __restrict__ q,
    float* __restrict__ scores)
{
  __shared__ float sW1p[2048];
  __shared__ float sW2p[4096];
  __shared__ float sB1[64], sB2[64], sQm[64];
  __shared__ float sHs[8 * 16 * HS_STRIDE];

  stage_weights(W1, b1, W2, b2, sW1p, sW2p, sB1, sB2);
  if (threadIdx.x < 64)  // mean over 2 heads * (1/sqrt(64))
    sQm[threadIdx.x] = (q[threadIdx.x] + q[64 + threadIdx.x]) * (0.5f * 0.125f);
  __syncthreads();

  const int wave = threadIdx.x >> 5;
  const int lane = threadIdx.x & 31;
  const int h = lane >> 4, r = lane & 15;
  const int b = blockIdx.y;
  const int row0 = blockIdx.x * 128 + wave * 16;
  float* hsb = sHs + wave * (16 * HS_STRIDE);

  Acc4 acc2 = mlp_tile16(feats + ((size_t)b * S_ + row0) * DIN,
                         sW1p, sW2p, sB1, sB2, hsb);

  // spill Hs2 tile and reduce each row against the mean query
#pragma unroll
  for (int nt = 0; nt < 4; ++nt)
#pragma unroll
    for (int v = 0; v < 8; ++v)
      hsb[(v + 8 * h) * HS_STRIDE + 16 * nt + r] = acc2.a[nt][v];

  if (lane < 16) {
    float s = 0.f;
#pragma unroll
    for (int c = 0; c < 64; ++c)
      s = fmaf(hsb[lane * HS_STRIDE + c], sQm[c], s);
    scores[(size_t)b * S_ + row0 + lane] = s;
  }
}

// ---------------------------------------------------------------------------
// Kernel 2: per-batch masked softmax + deterministic top-K. grid = B, 256 thr.
// ---------------------------------------------------------------------------
__global__ __launch_bounds__(256) void k_softmax_topk(
    const float* __restrict__ scores, const int* __restrict__ mask,
    float* __restrict__ attnW, int* __restrict__ topk)
{
  __shared__ float sl[S_];       // logits, later mutated for top-K
  __shared__ float red[256];
  __shared__ int   redi[256];
  __shared__ int   sAny;

  const int b = blockIdx.x;
  const int t = threadIdx.x;
  const float NEG_INF = -__builtin_inff();

  if (t == 0) sAny = 0;
  __syncthreads();

  int any = 0;
  for (int s = t; s < S_; s += 256) {
    int m = mask[(size_t)b * S_ + s];            // bool mask (integer input)
    any |= (m != 0);
    float sc = scores[(size_t)b * S_ + s];
    sl[s] = m ? sc : NEG_INF;
  }
  if (any) atomicOr(&sAny, 1);
  __syncthreads();

  const bool anyb = (sAny != 0);
  if (!anyb) {  // all-masked fallback: logits = 0 -> uniform softmax
    for (int s = t; s < S_; s += 256) sl[s] = 0.f;
  }
  __syncthreads();

  // ---- row max ----
  float mx = NEG_INF;
  for (int s = t; s < S_; s += 256) mx = fmaxf(mx, sl[s]);
  red[t] = mx; __syncthreads();
  for (int off = 128; off; off >>= 1) {
    if (t < off) red[t] = fmaxf(red[t], red[t + off]);
    __syncthreads();
  }
  mx = red[0]; __syncthreads();

  // ---- exp-sum ----
  float ls = 0.f;
  for (int s = t; s < S_; s += 256) ls += expf(sl[s] - mx);
  red[t] = ls; __syncthreads();
  for (int off = 128; off; off >>= 1) {
    if (t < off) red[t] += red[t + off];
    __syncthreads();
  }
  const float inv = 1.f / red[0];
  for (int s = t; s < S_; s += 256)
    attnW[(size_t)b * S_ + s] = expf(sl[s] - mx) * inv;
  __syncthreads();

  // ---- top-K: 8 argmax passes, stable (lowest index wins ties) ----
  for (int k = 0; k < K_; ++k) {
    float bv = NEG_INF; int bi = 0x7fffffff;
    for (int s = t; s < S_; s += 256) {
      float v = sl[s];
      if (v > bv || (v == bv && s < bi)) { bv = v; bi = s; }
    }
    red[t] = bv; redi[t] = bi; __syncthreads();
    for (int off = 128; off; off >>= 1) {
      if (t < off) {
        float ov = red[t + off]; int oi = redi[t + off];
        if (ov > red[t] || (ov == red[t] && oi < redi[t])) { red[t] = ov; redi[t] = oi; }
      }
      __syncthreads();
    }
    if (t == 0) { topk[b * K_ + k] = redi[0]; sl[redi[0]] = NEG_INF; }
    __syncthreads();
  }
}

// ---------------------------------------------------------------------------
// Kernel 3: recompute Hs tiles (WMMA) and form fixed-order ctx partials.
// grid = (S/128, B). No atomics -> bitwise deterministic.
// ---------------------------------------------------------------------------
__global__ __launch_bounds__(256) void k_ctx_partial(
    const float* __restrict__ feats, const float* __restrict__ W1,
    const float* __restrict__ b1,    const float* __restrict__ W2,
    const float* __restrict__ b2,    const float* __restrict__ attnW,
    float* __restrict__ part)        // [B][S/128][64]
{
  __shared__ float sW1p[2048];
  __shared__ float sW2p[4096];
  __shared__ float sB1[64], sB2[64];
  __shared__ float sHs[8 * 16 * HS_STRIDE];
  __shared__ float sWr[8][16];
  __shared__ float sPart[8][2][64];

  stage_weights(W1, b1, W2, b2, sW1p, sW2p, sB1, sB2);
  __syncthreads();

  const int wave = threadIdx.x >> 5;
  const int lane = threadIdx.x & 31;
  const int h = lane >> 4, r = lane & 15;
  const int b = blockIdx.y;
  const int row0 = blockIdx.x * 128 + wave * 16;
  float* hsb = sHs + wave * (16 * HS_STRIDE);

  Acc4 acc2 = mlp_tile16(feats + ((size_t)b * S_ + row0) * DIN,
                         sW1p, sW2p, sB1, sB2, hsb);

  if (lane < 16)
    sWr[wave][lane] = attnW[(size_t)b * S_ + row0 + lane];

  // lane (h, r) holds rows {v+8h} at columns {16*nt + r}
#pragma unroll
  for (int nt = 0; nt < 4; ++nt) {
    float p = 0.f;
#pragma unroll
    for (int v = 0; v < 8; ++v)
      p = fmaf(acc2.a[nt][v], sWr[wave][v + 8 * h], p);
    sPart[wave][h][16 * nt + r] = p;
  }
  __syncthreads();

  if (threadIdx.x < 64) {
    float s = 0.f;
#pragma unroll
    for (int w = 0; w < 8; ++w)
      s += sPart[w][0][threadIdx.x] + sPart[w][1][threadIdx.x];
    part[((size_t)b * (S_ / 128) + blockIdx.x) * 64 + threadIdx.x] = s;
  }
}

// Kernel 3b: fixed-order tile reduction -> ctx. grid = B, 64 threads.
__global__ __launch_bounds__(64) void k_ctx_reduce(
    const float* __restrict__ part, float* __restrict__ ctx)
{
  const int b = blockIdx.x, t = threadIdx.x;
  float s = 0.f;
#pragma unroll
  for (int tile = 0; tile < S_ / 128; ++tile)
    s += part[((size_t)b * (S_ / 128) + tile) * 64 + t];
  ctx[(size_t)b * 64 + t] = s;
}

// ---------------------------------------------------------------------------
// Kernel 4: gather sel = Hs[topk] by recomputing 8 rows/batch. grid = B*K, 64 thr.
// ---------------------------------------------------------------------------
__global__ __launch_bounds__(64) void k_sel_gather(
    const float* __restrict__ feats, const float* __restrict__ W1,
    const float* __restrict__ b1,    const float* __restrict__ W2,
    const float* __restrict__ b2,    const int* __restrict__ topk,
    float* __restrict__ sel)
{
  const int bk = blockIdx.x;
  const int b = bk >> 3;
  const int idx = topk[bk];
  const int t = threadIdx.x;

  __shared__ float sx[DIN];
  __shared__ float sh1[DSL];

  if (t < DIN) sx[t] = feats[((size_t)b * S_ + idx) * DIN + t];
  __syncthreads();

  float a = b1[t];
#pragma unroll
  for (int i = 0; i < DIN; ++i) a = fmaf(sx[i], W1[i * DSL + t], a);
  sh1[t] = fmaxf(a, 0.f);
  __syncthreads();

  float o = b2[t];
#pragma unroll
  for (int i = 0; i < DSL; ++i) o = fmaf(sh1[i], W2[i * DSL + t], o);
  sel[(size_t)bk * DSL + t] = o;
}

// ---------------------------------------------------------------------------
extern "C" void kernel_launch(void* const* d_in, const int* in_sizes, int n_in,
                              void* d_out, int out_size, void* d_ws, size_t ws_size,
                              hipStream_t stream) {
  const float* feats = (const float*)d_in[0];   // [B,S,32]
  const int*   mask  = (const int*)  d_in[1];   // [B,S] bool->int
  const float* W1    = (const float*)d_in[2];   // [32,64]
  const float* b1    = (const float*)d_in[3];   // [64]
  const float* W2    = (const float*)d_in[4];   // [64,64]
  const float* b2    = (const float*)d_in[5];   // [64]
  const float* q     = (const float*)d_in[6];   // [2,64]

  // outputs concatenated: sel [B,K,64] | ctx [B,64] | attnW [B,S]
  float* sel   = (float*)d_out;
  float* ctx   = sel + (size_t)B_ * K_ * DSL;
  float* attnW = ctx + (size_t)B_ * DSL;

  // workspace: scores | topk | ctx partials
  float* scores = (float*)d_ws;
  int*   topk   = (int*)((char*)d_ws + (size_t)B_ * S_ * sizeof(float));
  float* part   = (float*)((char*)topk + (size_t)B_ * K_ * sizeof(int));

  k_mlp_scores<<<dim3(S_ / 128, B_), 256, 0, stream>>>(feats, W1, b1, W2, b2, q, scores);
  k_softmax_topk<<<B_, 256, 0, stream>>>(scores, mask, attnW, topk);
  k_ctx_partial<<<dim3(S_ / 128, B_), 256, 0, stream>>>(feats, W1, b1, W2, b2, attnW, part);
  k_ctx_reduce<<<B_, 64, 0, stream>>>(part, ctx);
  k_sel_gather<<<B_ * K_, 64, 0, stream>>>(feats, W1, b1, W2, b2, topk, sel);
}